// PairwiseProjector_67877663146048
// MI455X (gfx1250) — compile-verified
//
#include <hip/hip_runtime.h>
#include <hip/hip_bf16.h>

#define NB 32
#define NP 4096
#define IH 480
#define IW 640
#define EPSF 1e-12f

typedef __attribute__((ext_vector_type(2))) float v2f;
typedef __attribute__((ext_vector_type(8))) float v8f;

// -------------------------------------------------------------------------
// Kernel 1: P[i][j] (3x4) = K * ( R_j*R_i^T | t_j - R_j*R_i^T*t_i )
// One thread per (i,j) pair. 1024 threads total.
// -------------------------------------------------------------------------
__global__ void build_P(const float* __restrict__ poses, const float* __restrict__ K,
                        float* __restrict__ P) {
  int tid = blockIdx.x * blockDim.x + threadIdx.x;
  if (tid >= NB * NB) return;
  int i = tid / NB;   // E_inv index (source batch)
  int j = tid % NB;   // E index (target camera)
  const float* pj = poses + j * 12;
  const float* pi = poses + i * 12;
  float Rj[3][3], Ri[3][3], tj[3], ti[3];
  for (int r = 0; r < 3; ++r) {
    for (int c = 0; c < 3; ++c) { Rj[r][c] = pj[r*4+c]; Ri[r][c] = pi[r*4+c]; }
    tj[r] = pj[r*4+3]; ti[r] = pi[r*4+3];
  }
  // M = R_j * R_i^T
  float M[3][3];
  for (int r = 0; r < 3; ++r)
    for (int c = 0; c < 3; ++c)
      M[r][c] = Rj[r][0]*Ri[c][0] + Rj[r][1]*Ri[c][1] + Rj[r][2]*Ri[c][2];
  float Mt[3];
  for (int r = 0; r < 3; ++r)
    Mt[r] = tj[r] - (M[r][0]*ti[0] + M[r][1]*ti[1] + M[r][2]*ti[2]);
  float Kl[3][3];
  for (int r = 0; r < 3; ++r)
    for (int c = 0; c < 3; ++c) Kl[r][c] = K[r*3+c];
  float* out = P + (size_t)tid * 12;   // row-major 3x4
  for (int r = 0; r < 3; ++r) {
    for (int c = 0; c < 3; ++c)
      out[r*4+c] = Kl[r][0]*M[0][c] + Kl[r][1]*M[1][c] + Kl[r][2]*M[2][c];
    out[r*4+3]  = Kl[r][0]*Mt[0]  + Kl[r][1]*Mt[1]  + Kl[r][2]*Mt[2];
  }
}

// -------------------------------------------------------------------------
// Kernel 2: cam[i][n] = Kinv * [px,py,1] * depth_bilinear ; stored [x,y,z,1]
// grid_sample: bilinear, padding_mode='zeros', align_corners=False semantics.
// -------------------------------------------------------------------------
__global__ void build_cam(const float* __restrict__ points, const float* __restrict__ depthsD,
                          const float* __restrict__ K, float* __restrict__ cam) {
  int tid = blockIdx.x * blockDim.x + threadIdx.x;
  if (tid >= NB * NP) return;
  int i = tid / NP;
  float gx = points[(size_t)tid*2 + 0];
  float gy = points[(size_t)tid*2 + 1];

  // grid_sample pixel coords
  float x = ((gx + 1.0f) * (float)IW - 1.0f) * 0.5f;
  float y = ((gy + 1.0f) * (float)IH - 1.0f) * 0.5f;
  float x0 = floorf(x), y0 = floorf(y);
  float wx1 = x - x0, wy1 = y - y0;
  const float* img = depthsD + (size_t)i * IH * IW;
  auto g = [&](float xf, float yf) -> float {
    bool valid = (xf >= 0.0f) && (xf <= (float)(IW-1)) && (yf >= 0.0f) && (yf <= (float)(IH-1));
    int xc = (int)fminf(fmaxf(xf, 0.0f), (float)(IW-1));
    int yc = (int)fminf(fmaxf(yf, 0.0f), (float)(IH-1));
    float v = img[(size_t)yc * IW + xc];
    return valid ? v : 0.0f;
  };
  float depth = g(x0,      y0     ) * (1.0f-wx1) * (1.0f-wy1)
              + g(x0+1.0f, y0     ) * wx1        * (1.0f-wy1)
              + g(x0,      y0+1.0f) * (1.0f-wx1) * wy1
              + g(x0+1.0f, y0+1.0f) * wx1        * wy1;

  // points_px = (points + 1) * [(W-1)/2, (H-1)/2]
  float px = (gx + 1.0f) * ((float)(IW-1) * 0.5f);
  float py = (gy + 1.0f) * ((float)(IH-1) * 0.5f);

  // analytic 3x3 inverse of K (adjugate)
  float a=K[0],b=K[1],c=K[2],d=K[3],e=K[4],f=K[5],g0=K[6],h=K[7],k=K[8];
  float A0 = e*k - f*h, A1 = -(d*k - f*g0), A2 = d*h - e*g0;
  float inv = 1.0f / (a*A0 + b*A1 + c*A2);
  float i00 = A0*inv, i01 = -(b*k - c*h)*inv, i02 =  (b*f - c*e)*inv;
  float i10 = A1*inv, i11 =  (a*k - c*g0)*inv, i12 = -(a*f - c*d)*inv;
  float i20 = A2*inv, i21 = -(a*h - b*g0)*inv, i22 =  (a*e - b*d)*inv;

  float cx = (i00*px + i01*py + i02) * depth;
  float cy = (i10*px + i11*py + i12) * depth;
  float cz = (i20*px + i21*py + i22) * depth;
  float4* o = (float4*)(cam + (size_t)tid * 4);
  *o = make_float4(cx, cy, cz, 1.0f);
}

// -------------------------------------------------------------------------
// Kernel 3: per wave = (batch i, 16-point tile); loop j over cams in pairs.
//   A(16x4): rows 0..2 = P[i,j], rows 8..10 = P[i,j+1]  (from LDS)
//   B(4x16): 16 homogeneous cam points [x,y,z,1]
//   D(16x16): d[0..2] = (u,v,z) in ALL lanes:
//     lanes 0..15  -> camera j,   point = lane
//     lanes 16..31 -> camera j+1, point = lane-16  (D rows 8..10)
// Full-wave epilogue: v_rcp, scale, boundary, coalesced float2 stores.
// -------------------------------------------------------------------------
__global__ void project_wmma(const float* __restrict__ camv, const float* __restrict__ Pm,
                             float* __restrict__ proj, float* __restrict__ bnd) {
  __shared__ float sP[NB * 12];                    // all 32 P matrices of batch i
  const int lane = threadIdx.x & 31;
  const int widx = threadIdx.x >> 5;               // 8 waves per block
  const int i    = blockIdx.x >> 5;                // 32 blocks per batch (exact)
  const int tile = ((blockIdx.x & 31) << 3) + widx;// 0..255

  // stage P[i][0..31] (384 floats = 1.5 KB) into LDS
  for (int t = threadIdx.x; t < NB * 12; t += blockDim.x)
    sP[t] = Pm[(size_t)i * NB * 12 + t];
  __syncthreads();

  const int row    = lane & 15;                    // A row (mod 8 halves) / point idx
  const int n      = tile * 16 + row;              // point index within batch i
  const int colOff = (lane < 16) ? 0 : 2;          // A/B: K=0,1 vs K=2,3 halves

  // B operand: column n = [camx,camy] (lo lanes) or [camz,1] (hi lanes)
  const float* c4 = camv + ((size_t)i * NP + n) * 4;
  v2f bmat;
  { const float2 t = *(const float2*)(c4 + colOff); bmat[0] = t.x; bmat[1] = t.y; }

  // branch-free A-fetch setup (per-lane constants)
  const int  hiRow = (row >= 8) ? 1 : 0;           // rows 8..10 -> camera j+1
  const int  r3    = row - 8 * hiRow;              // 0..7
  const bool avalid = (r3 < 3);
  const int  rc    = avalid ? r3 : 0;              // clamp for in-bounds LDS read
  const int  aoff  = hiRow * 12 + rc * 4 + colOff; // offset within camera-pair slab

  const float sxi = 1.0f / ((float)(IW-1) * 0.5f);
  const float syi = 1.0f / ((float)(IH-1) * 0.5f);
  const int   jj_off = (lane < 16) ? 0 : 1;

  for (int j = 0; j < NB; j += 2) {
    // A operand from LDS (ds_load_b64), zeroed outside rows 0..2 / 8..10
    const float2 av = *(const float2*)(sP + j * 12 + aoff);
    v2f amat;
    amat[0] = avalid ? av.x : 0.0f;
    amat[1] = avalid ? av.y : 0.0f;

    v8f cacc = {};
    v8f dmat = __builtin_amdgcn_wmma_f32_16x16x4_f32(
        /*neg_a=*/false, amat, /*neg_b=*/false, bmat,
        /*c_mod=*/(short)0, cacc, /*reuse_a=*/false, /*reuse_b=*/false);

    // full-wave epilogue: lanes 0..15 = cam j, lanes 16..31 = cam j+1
    float u = dmat[0], v = dmat[1], z = dmat[2];
    float zi = __builtin_amdgcn_rcpf(z + EPSF);    // v_rcp_f32 (~1 ulp)
    float pxo = (u * zi) * sxi - 1.0f;
    float pyo = (v * zi) * syi - 1.0f;
    const int jj = j + jj_off;
    size_t base = (((size_t)i * NB + jj) * NP + n);
    *(float2*)(proj + base * 2) = make_float2(pxo, pyo);
    bnd[base] = (pxo >= -1.0f && pxo <= 1.0f && pyo >= -1.0f && pyo <= 1.0f) ? 1.0f : 0.0f;
  }
}

// -------------------------------------------------------------------------
extern "C" void kernel_launch(void* const* d_in, const int* in_sizes, int n_in,
                              void* d_out, int out_size, void* d_ws, size_t ws_size,
                              hipStream_t stream) {
  const float* points = (const float*)d_in[0];   // (B,N,2)
  const float* depths = (const float*)d_in[1];   // (B,1,H,W)
  const float* poses  = (const float*)d_in[2];   // (B,3,4)
  const float* K      = (const float*)d_in[3];   // (3,3)

  float* camv = (float*)d_ws;                        // NB*NP*4 floats (~2 MB)
  float* Pm   = camv + (size_t)NB * NP * 4;          // NB*NB*12 floats (~48 KB)

  float* proj = (float*)d_out;                       // (B,B,N,2)
  float* bnd  = proj + (size_t)NB * NB * NP * 2;     // (B,B,N)

  hipLaunchKernelGGL(build_P, dim3((NB*NB + 255) / 256), dim3(256), 0, stream,
                     poses, K, Pm);
  hipLaunchKernelGGL(build_cam, dim3((NB*NP + 255) / 256), dim3(256), 0, stream,
                     points, depths, K, camv);
  const int waves = NB * (NP / 16);                  // 8192 waves
  hipLaunchKernelGGL(project_wmma, dim3(waves / 8), dim3(256), 0, stream,
                     camv, Pm, proj, bnd);
}